// GCNNet_65498251264053
// MI455X (gfx1250) — compile-verified
//
#include <hip/hip_runtime.h>
#include <hip/hip_bf16.h>

#define N_NODES  50000
#define N_EDGES  800000
#define D_FEAT   96
#define D_HID    10
#define N_GRAPHS 512

typedef float v2f __attribute__((ext_vector_type(2)));
typedef float v8f __attribute__((ext_vector_type(8)));

// ---------------------------------------------------------------------------
// Kernel 0: zero the workspace accumulators (g_sums[512*96] ++ counts[512]).
// Harness poisons ws once with 0xAA and never re-poisons, so we must zero on
// every call (we accumulate with atomics).
// ---------------------------------------------------------------------------
__global__ void gcn_zero_ws(float* __restrict__ ws, int n) {
    int i = blockIdx.x * blockDim.x + threadIdx.x;
    int stride = gridDim.x * blockDim.x;
    for (; i < n; i += stride) ws[i] = 0.0f;
}

// ---------------------------------------------------------------------------
// Kernel 1: counts[g] = # nodes with batch==g  (as float, for the divide).
// ---------------------------------------------------------------------------
__global__ void gcn_count_nodes(const int* __restrict__ batch,
                                float* __restrict__ counts, int n) {
    int i = blockIdx.x * blockDim.x + threadIdx.x;
    int stride = gridDim.x * blockDim.x;
    for (; i < n; i += stride) {
        atomicAdd(&counts[batch[i]], 1.0f);
    }
}

// ---------------------------------------------------------------------------
// Kernel 2: fused edge-message + node-aggregation + graph-pool (sum part).
//   g_sums[g][f] += edge_attr[e] * x[src[e]][f]   where g = batch[dst[e]]
// The 50000x96 node intermediate h is never materialized: the scatter target
// is only 512x96 floats. Each block accumulates a 32-feature slice
// (blockIdx.y in {0,1,2}) of all 512 graphs in a 64KB LDS tile via
// ds_add_f32, then merges into global with f32 atomics.
// One wave per edge; lane = feature within slice -> 128B coalesced x gather
// (x is 19MB, L2-resident on the 192MB L2 after first touch).
// ---------------------------------------------------------------------------
__global__ __launch_bounds__(256) void gcn_edge_scatter(
        const float* __restrict__ x,
        const int*   __restrict__ edge_index,   // [2*E], src then dst
        const float* __restrict__ edge_attr,
        const int*   __restrict__ batch,
        float* __restrict__ g_sums) {
    __shared__ float acc[N_GRAPHS * 32];        // 64 KB

    const int fbase = blockIdx.y * 32;          // feature slice base
    // zero the LDS accumulator
    for (int i = threadIdx.x; i < N_GRAPHS * 32; i += blockDim.x)
        acc[i] = 0.0f;
    __syncthreads();

    const int lane           = threadIdx.x & 31;
    const int wave_in_block  = threadIdx.x >> 5;
    const int waves_per_blk  = blockDim.x >> 5;
    const int gwave          = blockIdx.x * waves_per_blk + wave_in_block;
    const int nwaves         = gridDim.x * waves_per_blk;

    for (int e = gwave; e < N_EDGES; e += nwaves) {
        const int   src = edge_index[e];
        const int   dst = edge_index[N_EDGES + e];
        const float w   = edge_attr[e];
        const int   g   = batch[dst];
        const float v   = w * x[src * D_FEAT + fbase + lane];
        // lane -> distinct LDS bank for any g: bank = (g*32+lane) % 64
        atomicAdd(&acc[g * 32 + lane], v);
    }
    __syncthreads();

    // merge block-partial into global
    for (int i = threadIdx.x; i < N_GRAPHS * 32; i += blockDim.x) {
        const int g = i >> 5;
        const int f = fbase + (i & 31);
        atomicAdd(&g_sums[g * D_FEAT + f], acc[i]);
    }
}

// ---------------------------------------------------------------------------
// Kernel 3: pooled = relu(g_sums / max(counts,1));
//           z1 = relu(pooled @ W1 + b1);  out = z1 @ W2 + b2
// 512x96 @ 96x10 done with V_WMMA_F32_16X16X4_F32 (full f32).
// One wave (32 threads) per 16-row tile of graphs; 24 K-steps of K=4.
// W1 is staged ONCE into LDS, transposed and zero-padded to 16 columns, so
// the inner loop has no divergent (lm < 10) predication: the B-frag is a
// single aligned ds_load_b64 and the A-frag a single global_load_b64.
// A-frag (16x4 f32, ISA layout): lanes 0-15 hold {K0,K1} of row M=lane,
//                                lanes 16-31 hold {K2,K3} of row M=lane-16.
// B-frag (4x16):   lanes 0-15 hold {B[K0][n],B[K1][n]} for n=lane,
//                  lanes 16-31 hold {B[K2][n],B[K3][n]} for n=lane-16.
// D (16x16 f32, 8 VGPRs): vgpr r = row (r + 8*hi), col = lane&15.
// EXEC is all-ones at every WMMA (blockDim=32, no divergence in the loop).
// ---------------------------------------------------------------------------
#define W1T_STRIDE 100   // even (b64-aligned), not a multiple of 32 (banks)

__global__ __launch_bounds__(32) void gcn_pool_mlp(
        const float* __restrict__ g_sums,
        const float* __restrict__ counts,
        const float* __restrict__ W1,   // [96][10] row-major
        const float* __restrict__ b1,   // [10]
        const float* __restrict__ W2,   // [10]
        const float* __restrict__ b2,   // [1]
        float* __restrict__ out) {      // [512]
    const int m0   = blockIdx.x * 16;
    const int lane = threadIdx.x;       // 0..31
    const int lm   = lane & 15;
    const int hi   = lane >> 4;         // 0: K0/K1 , 1: K2/K3

    // ---- stage W1^T (zero-padded N=16) into LDS once -----------------------
    __shared__ float w1t[16 * W1T_STRIDE];      // w1t[n][k] = W1[k][n], 6.4KB
    for (int i = lane; i < 16 * D_FEAT; i += 32) {
        const int n = i / D_FEAT;
        const int k = i - n * D_FEAT;
        w1t[n * W1T_STRIDE + k] = (n < D_HID) ? W1[k * D_HID + n] : 0.0f;
    }
    __syncthreads();

    const int   rowA = m0 + lm;                       // graph row this lane feeds
    const float cnt  = counts[rowA];
    const float inv  = 1.0f / fmaxf(cnt, 1.0f);

    const v2f* __restrict__ arow = (const v2f*)(g_sums + rowA * D_FEAT);
    const v2f* __restrict__ brow = (const v2f*)(w1t + lm * W1T_STRIDE);

    v8f c = {};
    #pragma unroll 4
    for (int k = 0; k < D_FEAT; k += 4) {
        const int ka = k + hi * 2;                // even -> v2f index ka/2
        v2f a = arow[ka >> 1];                    // one global_load_b64
        a.x = fmaxf(a.x * inv, 0.0f);
        a.y = fmaxf(a.y * inv, 0.0f);
        v2f b = brow[ka >> 1];                    // one ds_load_b64
        c = __builtin_amdgcn_wmma_f32_16x16x4_f32(
                /*neg_a=*/false, a, /*neg_b=*/false, b,
                /*c_mod=*/(short)0, c, /*reuse_a=*/false, /*reuse_b=*/false);
    }

    // layer-1 bias + relu, stash z1 tile in LDS for the 10->1 reduction
    __shared__ float z1[16][17];
    const float bn = (lm < D_HID) ? b1[lm] : 0.0f;
    #pragma unroll
    for (int r = 0; r < 8; ++r) {
        float z = fmaxf(c[r] + bn, 0.0f);
        z1[r + 8 * hi][lm] = (lm < D_HID) ? z : 0.0f;
    }
    __syncthreads();

    if (lane < 16) {
        float s = b2[0];
        #pragma unroll
        for (int n = 0; n < D_HID; ++n) s += z1[lane][n] * W2[n];
        out[m0 + lane] = s;
    }
}

// ---------------------------------------------------------------------------
// Host-side launcher (graph-capture safe: only kernel launches on `stream`).
// ws layout: [0, 512*96) = g_sums ; [512*96, 512*96+512) = counts
// ---------------------------------------------------------------------------
extern "C" void kernel_launch(void* const* d_in, const int* in_sizes, int n_in,
                              void* d_out, int out_size, void* d_ws, size_t ws_size,
                              hipStream_t stream) {
    const float* x          = (const float*)d_in[0];   // [50000*96]
    const int*   edge_index = (const int*)  d_in[1];   // [2*800000]
    const float* edge_attr  = (const float*)d_in[2];   // [800000]
    const int*   batch      = (const int*)  d_in[3];   // [50000]
    const float* W1         = (const float*)d_in[4];   // [96*10]
    const float* b1         = (const float*)d_in[5];   // [10]
    const float* W2         = (const float*)d_in[6];   // [10]
    const float* b2         = (const float*)d_in[7];   // [1]
    float* out = (float*)d_out;                        // [512]

    float* g_sums = (float*)d_ws;                      // 512*96 floats
    float* counts = g_sums + N_GRAPHS * D_FEAT;        // 512 floats

    const int n_ws_floats = N_GRAPHS * D_FEAT + N_GRAPHS;
    gcn_zero_ws<<<(n_ws_floats + 255) / 256, 256, 0, stream>>>(g_sums, n_ws_floats);

    gcn_count_nodes<<<(N_NODES + 255) / 256, 256, 0, stream>>>(batch, counts, N_NODES);

    dim3 egrid(96, D_FEAT / 32);   // 96 edge-chunks x 3 feature slices
    gcn_edge_scatter<<<egrid, 256, 0, stream>>>(x, edge_index, edge_attr, batch, g_sums);

    gcn_pool_mlp<<<N_GRAPHS / 16, 32, 0, stream>>>(g_sums, counts, W1, b1, W2, b2, out);
}